// Transformer_Block_88244398063659
// MI455X (gfx1250) — compile-verified
//
#include <hip/hip_runtime.h>
#include <stdint.h>

// Problem constants from the reference
#define NN 50000
#define EE 800000
#define CC 64

typedef __attribute__((ext_vector_type(2))) float v2f;
typedef __attribute__((ext_vector_type(8))) float v8f;

// Guaranteed hardware f32 atomic add (no-return form -> STOREcnt, no data
// round-trip). Avoids any chance of a CAS retry loop for the 51.2M
// scatter-adds; these resolve in the L2 atomic units (whole working set is
// L2-resident on MI455X's 192 MB L2).
__device__ __forceinline__ void hw_atomic_add_f32(float* p, float v) {
    asm volatile("global_atomic_add_f32 %0, %1, off"
                 :
                 : "v"(p), "v"(v)
                 : "memory");
}

// ---------------------------------------------------------------------------
// Kernel 0: zero the output (harness poisons d_out with 0xAA).
// ---------------------------------------------------------------------------
__global__ __launch_bounds__(256) void zero_out_kernel(float4* __restrict__ p, int n4) {
    int i = blockIdx.x * 256 + threadIdx.x;
    if (i < n4) p[i] = make_float4(0.f, 0.f, 0.f, 0.f);
}

// ---------------------------------------------------------------------------
// Kernel 1: per 16-row tile of x:
//   x2  = x @ W2        (fp32 WMMA 16x16x4, K-loop of 16)
//   x4p = x @ W4
//   x3p = x * w3        (elementwise)
// 256 threads = 8 waves; wave w -> (matrix = w/4, ntile = w%4).
// N = 50000 = 3125 * 16 exactly, so no row remainder handling needed.
//
// WMMA f32 16x16x4 VGPR layout (ISA 7.12.2):
//   A (16x4): lanes 0-15: m=lane, {K0,K1}; lanes 16-31: m=lane-16, {K2,K3}
//   B (4x16): lanes 0-15: n=lane, {K0,K1}; lanes 16-31: n=lane-16, {K2,K3}
//   C/D (16x16): VGPR r: lanes 0-15 -> M=r, lanes 16-31 -> M=r+8; N=lane&15
// ---------------------------------------------------------------------------
__global__ __launch_bounds__(256) void gemm_tiles_kernel(
    const float* __restrict__ x,
    const float* __restrict__ W2,
    const float* __restrict__ w3,
    const float* __restrict__ W4,
    float* __restrict__ x2,
    float* __restrict__ x4p,
    float* __restrict__ x3p)
{
    // 16 rows x 64 cols, padded to stride 65 so column-wise reads
    // (fixed k, varying row) hit 16 distinct banks instead of one.
    __shared__ float lx[16 * 65];

    const int tid  = threadIdx.x;
    const int row0 = blockIdx.x * 16;

    // Cooperative load of the 16x64 x-tile (1024 floats, 4 per thread).
    #pragma unroll
    for (int i = 0; i < 4; ++i) {
        int idx = tid + i * 256;
        int r = idx >> 6, c = idx & 63;
        lx[r * 65 + c] = x[(size_t)row0 * CC + idx];
    }
    __syncthreads();

    // x3p = x * w3 while the tile is hot in LDS.
    #pragma unroll
    for (int i = 0; i < 4; ++i) {
        int idx = tid + i * 256;
        int r = idx >> 6, c = idx & 63;
        x3p[(size_t)row0 * CC + idx] = lx[r * 65 + c] * w3[c];
    }

    const int wave  = tid >> 5;
    const int lane  = tid & 31;
    const int ntile = wave & 3;
    const float* W    = (wave < 4) ? W2 : W4;
    float*       outm = (wave < 4) ? x2 : x4p;

    const int n   = lane & 15;              // N index (also M index for A frag)
    const int kb  = (lane < 16) ? 0 : 2;    // K sub-offset for this half-wave
    const int col = ntile * 16 + n;

    // Preload all 16 B fragments (W is 16 KB, L2-resident): 2 VGPRs each.
    v2f bfrag[16];
    #pragma unroll
    for (int ks = 0; ks < 16; ++ks) {
        int k = ks * 4 + kb;
        bfrag[ks].x = W[(size_t)k * CC + col];
        bfrag[ks].y = W[(size_t)(k + 1) * CC + col];
    }

    // K-loop: 16 x V_WMMA_F32_16X16X4_F32, A fragments streamed from LDS.
    v8f acc = {};
    #pragma unroll
    for (int ks = 0; ks < 16; ++ks) {
        int k = ks * 4 + kb;
        v2f afrag;
        afrag.x = lx[n * 65 + k];
        afrag.y = lx[n * 65 + k + 1];
        acc = __builtin_amdgcn_wmma_f32_16x16x4_f32(
            /*neg_a=*/false, afrag, /*neg_b=*/false, bfrag[ks],
            /*c_mod=*/(short)0, acc, /*reuse_a=*/false, /*reuse_b=*/false);
    }

    // Store D tile: VGPR r holds rows (r, r+8) split across lane halves.
    const int mbase = (lane < 16) ? 0 : 8;
    #pragma unroll
    for (int r = 0; r < 8; ++r) {
        outm[(size_t)(row0 + mbase + r) * CC + col] = acc[r];
    }
}

// ---------------------------------------------------------------------------
// Kernel 2: one wave (32 lanes) per edge.
//   w_e = <x3p[src,:], x4p[dst,:]>   (float2 per lane + 5-step xor butterfly)
//   out[dst,:] += w_e * x2[src,:]    (hardware f32 atomics into L2)
// All row gathers are 256 B contiguous per wave (float2 per lane).
// ---------------------------------------------------------------------------
__global__ __launch_bounds__(256) void edge_scatter_kernel(
    const long long* __restrict__ ei,   // [2, E] int64: row 0 = src, row 1 = dst
    const float* __restrict__ x2,
    const float* __restrict__ x3p,
    const float* __restrict__ x4p,
    float* __restrict__ out)
{
    const int lane = threadIdx.x & 31;
    const int e    = blockIdx.x * 8 + (threadIdx.x >> 5);
    if (e >= EE) return;

    const int src = (int)ei[e];
    const int dst = (int)ei[EE + e];

    const float2 a = ((const float2*)(x3p + (size_t)src * CC))[lane];
    const float2 b = ((const float2*)(x4p + (size_t)dst * CC))[lane];
    float p = a.x * b.x + a.y * b.y;

    // wave32 butterfly reduction -> every lane holds w_e
    #pragma unroll
    for (int off = 16; off > 0; off >>= 1)
        p += __shfl_xor(p, off, 32);

    const float2 c = ((const float2*)(x2 + (size_t)src * CC))[lane];
    float* o = out + (size_t)dst * CC + 2 * lane;
    hw_atomic_add_f32(o,     p * c.x);
    hw_atomic_add_f32(o + 1, p * c.y);
}

// ---------------------------------------------------------------------------
// Launch: inputs in setup_inputs() order: x, edge_index, W2, w3, W4.
// Workspace layout: [x2 | x4p | x3p], each N*C fp32 (38.4 MB total).
// ---------------------------------------------------------------------------
extern "C" void kernel_launch(void* const* d_in, const int* in_sizes, int n_in,
                              void* d_out, int out_size, void* d_ws, size_t ws_size,
                              hipStream_t stream) {
    const float*     x   = (const float*)d_in[0];
    const long long* ei  = (const long long*)d_in[1];   // int64 per reference
    const float*     W2  = (const float*)d_in[2];
    const float*     w3  = (const float*)d_in[3];
    const float*     W4  = (const float*)d_in[4];
    float*           out = (float*)d_out;

    float* x2  = (float*)d_ws;
    float* x4p = x2  + (size_t)NN * CC;
    float* x3p = x4p + (size_t)NN * CC;

    // 1) zero output (N*C = 3.2M floats = 800k float4)
    zero_out_kernel<<<(NN * CC / 4 + 255) / 256, 256, 0, stream>>>(
        (float4*)out, NN * CC / 4);

    // 2) GEMMs + elementwise: 50000/16 = 3125 row tiles
    gemm_tiles_kernel<<<NN / 16, 256, 0, stream>>>(x, W2, w3, W4, x2, x4p, x3p);

    // 3) edge dot + scatter-add: 8 edges per 256-thread block
    edge_scatter_kernel<<<(EE + 7) / 8, 256, 0, stream>>>(ei, x2, x3p, x4p, out);
}